// GraphLayer_36232344109604
// MI455X (gfx1250) — compile-verified
//
#include <hip/hip_runtime.h>

#define N_NODES 100000
#define NE      1600000
#define DIM     128
#define NH      8
#define DH      16
#define NEG_SLOPE 0.2f
#define LN_EPS  1e-5f
#define APITCH  132   // padded LDS row pitch in floats (528 B, 16B-aligned, bank-conflict free)

typedef __attribute__((ext_vector_type(2))) float v2f;
typedef __attribute__((ext_vector_type(8))) float v8f;

// order-preserving float<->uint encoding for atomicMax on floats
__device__ __forceinline__ unsigned fenc(float f) {
    unsigned u = __float_as_uint(f);
    return (u & 0x80000000u) ? ~u : (u | 0x80000000u);
}
__device__ __forceinline__ float fdec(unsigned u) {
    return __uint_as_float((u & 0x80000000u) ? (u & 0x7fffffffu) : ~u);
}
__device__ __forceinline__ float lrelu(float v) { return v > 0.f ? v : NEG_SLOPE * v; }

// Async memory->LDS tile copy: 16 rows x 128 floats, 16B per lane-op, ASYNCcnt-tracked.
// Each wave issues exactly 2 global_load_async_to_lds_b128 per tile (512 chunks / 256 thr).
__device__ __forceinline__ void tile_async_load(float* lds_tile, const float* src,
                                                int row0, int tid) {
    #pragma unroll
    for (int t = tid; t < 512; t += 256) {
        int r  = t >> 5;        // row 0..15
        int c4 = t & 31;        // 16B chunk within row
        unsigned ldsa = (unsigned)(uintptr_t)(lds_tile + r * APITCH + c4 * 4);
        const float* g = src + (size_t)(row0 + r) * DIM + c4 * 4;
        asm volatile("global_load_async_to_lds_b128 %0, %1, off"
                     :: "v"(ldsa), "v"(g) : "memory");
    }
}

// ---------------------------------------------------------------------------
// Dual-source WMMA fp32 GEMM: out[16 rows x 128 cols per block]
//   C = A1@W1 (+ A2@W2) (+bias); optional epilogue: y = C + resid, LayerNorm(y)
// Block = 256 threads = 8 waves; wave w owns output cols [16w, 16w+16).
// A tiles staged via async-to-LDS (double buffered across the two passes);
// V_WMMA_F32_16X16X4_F32 (exact fp32) sweeps K four at a time.
// ---------------------------------------------------------------------------
__global__ __launch_bounds__(256) void gemm_dual_kernel(
    const float* __restrict__ A1, const float* __restrict__ W1,
    const float* __restrict__ A2, const float* __restrict__ W2,
    const float* __restrict__ bias, const float* __restrict__ resid,
    const float* __restrict__ gamma, const float* __restrict__ beta,
    float* __restrict__ out, int do_ln) {
    __shared__ float As[2][16 * APITCH];   // double-buffered A tiles
    __shared__ float Yt[16 * APITCH];      // LN staging tile

    const int tid  = threadIdx.x;
    const int wave = tid >> 5;
    const int lane = tid & 31;
    const int row0 = blockIdx.x * 16;
    const int m    = lane & 15;            // A row / C col-in-tile
    const int koff = (lane >= 16) ? 2 : 0; // upper half-wave carries K+2
    const int col  = wave * 16 + m;        // global output column (0..127)

    // Queue both tiles up front; pass-1 copy hides under pass-0 WMMA chain.
    tile_async_load(As[0], A1, row0, tid);
    const bool dual = (A2 != nullptr);
    if (dual) tile_async_load(As[1], A2, row0, tid);

    v8f c = {0.f, 0.f, 0.f, 0.f, 0.f, 0.f, 0.f, 0.f};

    if (dual) asm volatile("s_wait_asynccnt 0x2" ::: "memory");  // buf0 landed
    else      asm volatile("s_wait_asynccnt 0x0" ::: "memory");
    __syncthreads();

    const float* Wsrc[2] = {W1, W2};
    for (int p = 0; p < 2; ++p) {
        const float* W = Wsrc[p];
        const float* At = As[p];
        #pragma unroll 8
        for (int kk = 0; kk < DIM; kk += 4) {
            v2f a, b;
            a.x = At[m * APITCH + kk + koff];
            a.y = At[m * APITCH + kk + koff + 1];
            b.x = W[(size_t)(kk + koff) * DIM + col];
            b.y = W[(size_t)(kk + koff + 1) * DIM + col];
            c = __builtin_amdgcn_wmma_f32_16x16x4_f32(
                    false, a, false, b, (short)0, c, false, false);
        }
        if (!dual) break;
        if (p == 0) {
            asm volatile("s_wait_asynccnt 0x0" ::: "memory");    // buf1 landed (this wave)
            __syncthreads();                                     // ... and all other waves
        }
    }

    const float bcol = bias ? bias[col] : 0.f;
    const int rb = (lane < 16) ? 0 : 8;   // C VGPR r holds rows r (lo half) / r+8 (hi half)

    if (!do_ln) {
        #pragma unroll
        for (int r = 0; r < 8; ++r)
            out[(size_t)(row0 + rb + r) * DIM + col] = c[r] + bcol;
        return;
    }

    // epilogue: y = C + bias + resid, then per-row LayerNorm over 128
    #pragma unroll
    for (int r = 0; r < 8; ++r) {
        int row = rb + r;
        Yt[row * APITCH + col] = c[r] + bcol + resid[(size_t)(row0 + row) * DIM + col];
    }
    __syncthreads();
    #pragma unroll
    for (int rr = 0; rr < 2; ++rr) {
        int row = wave * 2 + rr;
        float s = 0.f, q = 0.f;
        #pragma unroll
        for (int k = 0; k < 4; ++k) {
            float v = Yt[row * APITCH + lane * 4 + k];
            s += v; q += v * v;
        }
        #pragma unroll
        for (int off = 16; off; off >>= 1) {
            s += __shfl_xor(s, off);
            q += __shfl_xor(q, off);
        }
        float mu  = s * (1.f / 128.f);
        float var = q * (1.f / 128.f) - mu * mu;
        float inv = rsqrtf(var + LN_EPS);
        #pragma unroll
        for (int k = 0; k < 4; ++k) {
            int cc = lane * 4 + k;
            float yv = Yt[row * APITCH + cc];
            out[(size_t)(row0 + row) * DIM + cc] = (yv - mu) * inv * gamma[cc] + beta[cc];
        }
    }
}

// ---------------------------------------------------------------------------
// Per-node init: attention coefficients, self-loop seeds, zero accumulators
// ---------------------------------------------------------------------------
__global__ void node_init_kernel(
    const float* __restrict__ xp,
    const float* __restrict__ att_src, const float* __restrict__ att_dst,
    float* __restrict__ as_, float* __restrict__ ad_,
    unsigned* __restrict__ emax_u, float* __restrict__ denom,
    float* __restrict__ cnt, float* __restrict__ gacc, float* __restrict__ agg) {
    int tid = blockIdx.x * blockDim.x + threadIdx.x;
    if (tid >= N_NODES * DIM) return;
    gacc[tid] = 0.f;
    agg[tid]  = 0.f;
    int i = tid >> 7, cc = tid & 127;
    if (cc < NH) {
        int h = cc;
        const float* xr = xp + (size_t)i * DIM + h * DH;
        float s = 0.f, d = 0.f;
        #pragma unroll
        for (int k = 0; k < DH; ++k) {
            float v = xr[k];
            s += v * att_src[h * DH + k];
            d += v * att_dst[h * DH + k];
        }
        as_[i * NH + h] = s;
        ad_[i * NH + h] = d;
        float es = lrelu(s + d);          // self-loop logit seeds the segment max
        emax_u[i * NH + h] = fenc(es);
        denom[i * NH + h]  = 0.f;
    }
    if (cc == 0) cnt[i] = 0.f;
}

// ---------------------------------------------------------------------------
// Edge pass 1: per-(edge,head) segment max via encoded-uint atomicMax
// ---------------------------------------------------------------------------
__global__ void edge_max_kernel(
    const int* __restrict__ src, const int* __restrict__ dst,
    const float* __restrict__ as_, const float* __restrict__ ad_,
    unsigned* __restrict__ emax_u) {
    int t = blockIdx.x * blockDim.x + threadIdx.x;
    if (t >= NE * NH) return;
    int e = t >> 3, h = t & 7;
    int s = src[e], d = dst[e];
    float ev = lrelu(as_[s * NH + h] + ad_[d * NH + h]);
    atomicMax(&emax_u[d * NH + h], fenc(ev));
}

// ---------------------------------------------------------------------------
// Edge pass 2 (fused GAT + SAGE): one wave32 per edge, 4 floats per lane.
// alpha = ee/denom is applied post-hoc, so one pass suffices.
// ---------------------------------------------------------------------------
__global__ __launch_bounds__(256) void edge_accum_kernel(
    const int* __restrict__ src, const int* __restrict__ dst,
    const float* __restrict__ as_, const float* __restrict__ ad_,
    const unsigned* __restrict__ emax_u,
    const float* __restrict__ xp, const float* __restrict__ x,
    float* __restrict__ denom, float* __restrict__ gacc,
    float* __restrict__ agg, float* __restrict__ cnt) {
    int wv   = (blockIdx.x * blockDim.x + threadIdx.x) >> 5;
    int lane = threadIdx.x & 31;
    if (wv >= NE) return;
    int s = src[wv], d = dst[wv];
    int j = lane * 4;      // column base 0..124
    int h = j >> 4;        // head owning these 4 columns
    float ev = lrelu(as_[s * NH + h] + ad_[d * NH + h]);
    float ee = expf(ev - fdec(emax_u[d * NH + h]));   // <= 1 by construction
    if ((lane & 3) == 0) atomicAdd(&denom[d * NH + h], ee);
    float4 xpv = ((const float4*)(xp + (size_t)s * DIM))[lane];
    float4 xv  = ((const float4*)(x  + (size_t)s * DIM))[lane];
    float* gd = gacc + (size_t)d * DIM + j;
    atomicAdd(gd + 0, xpv.x * ee); atomicAdd(gd + 1, xpv.y * ee);
    atomicAdd(gd + 2, xpv.z * ee); atomicAdd(gd + 3, xpv.w * ee);
    float* ag = agg + (size_t)d * DIM + j;
    atomicAdd(ag + 0, xv.x); atomicAdd(ag + 1, xv.y);
    atomicAdd(ag + 2, xv.z); atomicAdd(ag + 3, xv.w);
    if (lane == 0) atomicAdd(&cnt[d], 1.0f);
}

// ---------------------------------------------------------------------------
// Per-node finalize: add self-loop term, normalize softmax, SAGE mean (in place)
// ---------------------------------------------------------------------------
__global__ void node_finalize_kernel(
    const float* __restrict__ as_, const float* __restrict__ ad_,
    const unsigned* __restrict__ emax_u, const float* __restrict__ denom,
    const float* __restrict__ cnt, const float* __restrict__ xp,
    const float* __restrict__ b_gat,
    float* __restrict__ gacc /* -> gat_out */, float* __restrict__ agg /* -> mean */) {
    int tid = blockIdx.x * blockDim.x + threadIdx.x;
    if (tid >= N_NODES * DIM) return;
    int i = tid >> 7, cc = tid & 127, h = cc >> 4;
    float es  = lrelu(as_[i * NH + h] + ad_[i * NH + h]);
    float ee  = expf(es - fdec(emax_u[i * NH + h]));
    float den = denom[i * NH + h] + ee;
    gacc[tid] = (gacc[tid] + xp[tid] * ee) / den + b_gat[cc];
    agg[tid]  = agg[tid] / fmaxf(cnt[i], 1.0f);
}

// ---------------------------------------------------------------------------
extern "C" void kernel_launch(void* const* d_in, const int* in_sizes, int n_in,
                              void* d_out, int out_size, void* d_ws, size_t ws_size,
                              hipStream_t stream) {
    const float* x       = (const float*)d_in[0];
    const int*   ei      = (const int*)  d_in[1];
    const float* W_gat   = (const float*)d_in[2];
    const float* att_src = (const float*)d_in[3];
    const float* att_dst = (const float*)d_in[4];
    const float* b_gat   = (const float*)d_in[5];
    const float* W_l     = (const float*)d_in[6];
    const float* W_r     = (const float*)d_in[7];
    const float* b_sage  = (const float*)d_in[8];
    const float* W_proj  = (const float*)d_in[9];
    const float* b_proj  = (const float*)d_in[10];
    const float* gamma   = (const float*)d_in[11];
    const float* beta    = (const float*)d_in[12];
    const int* src = ei;
    const int* dst = ei + NE;
    float* outp = (float*)d_out;

    float* w = (float*)d_ws;
    float* xp    = w; w += (size_t)N_NODES * DIM;   // x @ W_gat
    float* gacc  = w; w += (size_t)N_NODES * DIM;   // GAT accumulator -> gat_out
    float* agg   = w; w += (size_t)N_NODES * DIM;   // SAGE sum -> mean
    float* sage  = w; w += (size_t)N_NODES * DIM;   // sage_out
    float* as_   = w; w += (size_t)N_NODES * NH;
    float* ad_   = w; w += (size_t)N_NODES * NH;
    unsigned* emax_u = (unsigned*)w; w += (size_t)N_NODES * NH;
    float* denom = w; w += (size_t)N_NODES * NH;
    float* cnt   = w; w += (size_t)N_NODES;

    const int gemm_blocks = (N_NODES + 15) / 16;       // 6250 (N % 16 == 0)
    const int node_elems  = N_NODES * DIM;

    // 1) xp = x @ W_gat
    gemm_dual_kernel<<<gemm_blocks, 256, 0, stream>>>(
        x, W_gat, nullptr, nullptr, nullptr, nullptr, nullptr, nullptr, xp, 0);
    // 2) attention coeffs + self-loop seed + zero accumulators
    node_init_kernel<<<(node_elems + 255) / 256, 256, 0, stream>>>(
        xp, att_src, att_dst, as_, ad_, emax_u, denom, cnt, gacc, agg);
    // 3) segment max over edges
    edge_max_kernel<<<(NE * NH + 255) / 256, 256, 0, stream>>>(
        src, dst, as_, ad_, emax_u);
    // 4) fused GAT exp-weighted scatter + SAGE sum/count (1 wave / edge)
    edge_accum_kernel<<<NE / 8, 256, 0, stream>>>(
        src, dst, as_, ad_, emax_u, xp, x, denom, gacc, agg, cnt);
    // 5) normalize softmax (+ self loop), SAGE mean
    node_finalize_kernel<<<(node_elems + 255) / 256, 256, 0, stream>>>(
        as_, ad_, emax_u, denom, cnt, xp, b_gat, gacc, agg);
    // 6) sage_out = mean @ W_l + x @ W_r + b_sage
    gemm_dual_kernel<<<gemm_blocks, 256, 0, stream>>>(
        agg, W_l, x, W_r, b_sage, nullptr, nullptr, nullptr, sage, 0);
    // 7) h = gat_out @ Wp_top + sage_out @ Wp_bot + b_proj; y = h + x; LayerNorm
    gemm_dual_kernel<<<gemm_blocks, 256, 0, stream>>>(
        gacc, W_proj, sage, W_proj + DIM * DIM, b_proj, x, gamma, beta, outp, 1);
}